// SNN_42125039239892
// MI455X (gfx1250) — compile-verified
//
#include <hip/hip_runtime.h>
#include <stdint.h>

#define THRESHOLD 1.0f
#define BETA1 0.9f
#define BETA2 0.8f

typedef __attribute__((ext_vector_type(2))) float v2f;
typedef __attribute__((ext_vector_type(8))) float v8f;

// D = A(16x4) * B(4x16) + C, fp32 in/out, wave32.
__device__ __forceinline__ v8f wmma_f32_16x16x4(v2f a, v2f b, v8f c) {
  return __builtin_amdgcn_wmma_f32_16x16x4_f32(false, a, false, b, (short)0, c,
                                               false, false);
}

// Uniform-base + 32-bit byte-offset load -> saddr-form global_load_b64.
__device__ __forceinline__ v2f ld2(const float* __restrict__ base,
                                   uint32_t byteOff) {
  return *(const v2f*)((const char*)base + (size_t)byteOff);
}

struct Frags {
  v2f a0, a1, b0, b1, b2, b3;
};

// -----------------------------------------------------------------------------
// NT GEMM: C[M,N] = A[M,K] @ B[N,K]^T   (A, B row-major, K contiguous in both).
// Wave tile 32(M) x 64(N): 2 M-subtiles x 4 N-subtiles of 16x16, K stepped by 4
// via v_wmma_f32_16x16x4_f32. Block: 256 threads = 8 waves, 4(M) x 2(N),
// block tile 128 x 128. M, N multiples of 128; K multiple of 8.
// Depth-2 software pipeline: loads issue two 8-WMMA groups ahead of use.
//
// fp32 WMMA operand layout (ISA 7.12.2):
//   A 16x4:  lanes 0-15 M=0..15 K={0,1}; lanes 16-31 M=0..15 K={2,3}
//   B 4x16:  VGPR0 lanes0-15 K=0 / lanes16-31 K=2; VGPR1 K=1 / K=3
// => both frags are a per-lane float2 at (row*K + k + 2*(lane>>4)).
// -----------------------------------------------------------------------------
__global__ __launch_bounds__(256) void gemm_nt_128x128(
    const float* __restrict__ A, const float* __restrict__ B,
    float* __restrict__ C, int M, int N, int K) {
  const int lane = threadIdx.x & 31;
  const int wave = threadIdx.x >> 5;
  const int l15  = lane & 15;
  const int half = lane >> 4;

  const int m0 = blockIdx.y * 128 + (wave & 3) * 32;
  const int n0 = blockIdx.x * 128 + (wave >> 2) * 64;

  // Per-lane 32-bit byte offsets; base pointers stay uniform (SGPR).
  const uint32_t oA0 = (uint32_t)(((m0 + l15) * K + 2 * half) * 4);
  const uint32_t oA1 = (uint32_t)(((m0 + 16 + l15) * K + 2 * half) * 4);
  const uint32_t oB0 = (uint32_t)(((n0 + l15) * K + 2 * half) * 4);
  const uint32_t oB1 = (uint32_t)(((n0 + 16 + l15) * K + 2 * half) * 4);
  const uint32_t oB2 = (uint32_t)(((n0 + 32 + l15) * K + 2 * half) * 4);
  const uint32_t oB3 = (uint32_t)(((n0 + 48 + l15) * K + 2 * half) * 4);

  auto load_group = [&](uint32_t kb) {
    Frags f;
    f.a0 = ld2(A, oA0 + kb);
    f.a1 = ld2(A, oA1 + kb);
    f.b0 = ld2(B, oB0 + kb);
    f.b1 = ld2(B, oB1 + kb);
    f.b2 = ld2(B, oB2 + kb);
    f.b3 = ld2(B, oB3 + kb);
    return f;
  };

  v8f acc[2][4];
  v8f z = {};
#pragma unroll
  for (int i = 0; i < 2; ++i)
#pragma unroll
    for (int s = 0; s < 4; ++s) acc[i][s] = z;

  auto compute = [&](const Frags& f) {
    acc[0][0] = wmma_f32_16x16x4(f.a0, f.b0, acc[0][0]);
    acc[1][0] = wmma_f32_16x16x4(f.a1, f.b0, acc[1][0]);
    acc[0][1] = wmma_f32_16x16x4(f.a0, f.b1, acc[0][1]);
    acc[1][1] = wmma_f32_16x16x4(f.a1, f.b1, acc[1][1]);
    acc[0][2] = wmma_f32_16x16x4(f.a0, f.b2, acc[0][2]);
    acc[1][2] = wmma_f32_16x16x4(f.a1, f.b2, acc[1][2]);
    acc[0][3] = wmma_f32_16x16x4(f.a0, f.b3, acc[0][3]);
    acc[1][3] = wmma_f32_16x16x4(f.a1, f.b3, acc[1][3]);
  };

  // Two groups in flight before the first consume.
  Frags f0 = load_group(0);
  Frags f1 = load_group(16);

#pragma unroll 4
  for (int k = 8; k < K; k += 4) {
    Frags fn = load_group((uint32_t)k * 4u);
    compute(f0);
    f0 = f1;
    f1 = fn;
  }
  compute(f0);
  compute(f1);

  // C/D 16x16 layout: VGPR r -> M = r (lanes 0-15) or r+8 (lanes 16-31),
  // N = colbase + (lane&15).
#pragma unroll
  for (int i = 0; i < 2; ++i) {
#pragma unroll
    for (int s = 0; s < 4; ++s) {
      const uint32_t col = (uint32_t)(n0 + s * 16 + l15);
#pragma unroll
      for (int r = 0; r < 8; ++r) {
        const uint32_t row = (uint32_t)(m0 + i * 16 + r + 8 * half);
        C[(size_t)row * (uint32_t)N + col] = acc[i][s][r];
      }
    }
  }
}

// -----------------------------------------------------------------------------
// Layer-1 LIF scan: parallel over (b,h), serial over the time chunk.
// Reads cur1 (GEMM output, pre-bias), overwrites it in place with spk1.
// Emits spk1_trace (batch row 0) and mem1_trace (element [0,0]).
// -----------------------------------------------------------------------------
__global__ __launch_bounds__(256) void lif_scan1(
    float* __restrict__ curbuf, const float* __restrict__ bias1,
    float* __restrict__ mem1, float* __restrict__ out_spk1,
    float* __restrict__ out_mem1, int Tc, int t0, int Bb, int NH) {
  const int h = blockIdx.x * blockDim.x + threadIdx.x;
  const int b = blockIdx.y;
  float m = mem1[(size_t)b * NH + h];
  const float bias = bias1[h];
  for (int t = 0; t < Tc; ++t) {
    const size_t idx = ((size_t)t * Bb + b) * NH + h;
    const float reset = (m > THRESHOLD) ? 1.0f : 0.0f;
    const float cur = curbuf[idx] + bias;
    m = BETA1 * m + cur - reset * THRESHOLD;
    const float spk = (m > THRESHOLD) ? 1.0f : 0.0f;
    curbuf[idx] = spk;  // in-place: cur1 -> spk1 (GEMM2's A operand)
    if (b == 0) {
      out_spk1[(size_t)(t0 + t) * NH + h] = spk;
      if (h == 0) out_mem1[t0 + t] = m;
    }
  }
  mem1[(size_t)b * NH + h] = m;
}

// -----------------------------------------------------------------------------
// Layer-2 LIF scan: parallel over (b,o), serial over the chunk; spk2 goes
// straight into d_out (it is the spk2_rec output).
// -----------------------------------------------------------------------------
__global__ __launch_bounds__(256) void lif_scan2(
    const float* __restrict__ curbuf, const float* __restrict__ bias2,
    float* __restrict__ mem2, float* __restrict__ out_spk2, int Tc, int t0,
    int Bb, int NO) {
  const int idx = blockIdx.x * blockDim.x + threadIdx.x;
  const int b = idx / NO;
  const int o = idx % NO;
  float m = mem2[(size_t)b * NO + o];
  const float bias = bias2[o];
  for (int t = 0; t < Tc; ++t) {
    const float reset = (m > THRESHOLD) ? 1.0f : 0.0f;
    const float cur = curbuf[((size_t)t * Bb + b) * NO + o] + bias;
    m = BETA2 * m + cur - reset * THRESHOLD;
    const float spk = (m > THRESHOLD) ? 1.0f : 0.0f;
    out_spk2[((size_t)(t0 + t) * Bb + b) * NO + o] = spk;
  }
  mem2[(size_t)b * NO + o] = m;
}

__global__ __launch_bounds__(256) void zero_kernel(float* __restrict__ p,
                                                   size_t n) {
  size_t i = (size_t)blockIdx.x * blockDim.x + threadIdx.x;
  if (i < n) p[i] = 0.0f;
}

extern "C" void kernel_launch(void* const* d_in, const int* in_sizes, int n_in,
                              void* d_out, int out_size, void* d_ws,
                              size_t ws_size, hipStream_t stream) {
  const float* x  = (const float*)d_in[0];   // (T, B, NI)
  const float* W1 = (const float*)d_in[1];   // (NH, NI)
  const float* b1 = (const float*)d_in[2];   // (NH)
  const float* W2 = (const float*)d_in[3];   // (NO, NH)
  const float* b2 = (const float*)d_in[4];   // (NO)
  (void)in_sizes; (void)n_in; (void)out_size;

  const int T = 64, Bb = 256, NI = 2048, NH = 4096, NO = 128;

  float* out = (float*)d_out;
  float* out_spk1 = out;                      // T*NH
  float* out_mem1 = out + (size_t)T * NH;     // T
  float* out_spk2 = out_mem1 + T;             // T*B*NO

  // Workspace: persistent state + time-chunk activation buffers.
  float* mem1 = (float*)d_ws;                     // B*NH
  float* mem2 = mem1 + (size_t)Bb * NH;           // B*NO
  float* cur1 = mem2 + (size_t)Bb * NO;           // Tc*B*NH (becomes spk1)
  const size_t state_elems = (size_t)Bb * NH + (size_t)Bb * NO;
  const size_t per_step = (size_t)Bb * NH + (size_t)Bb * NO;

  // Chunk size: largest power-of-two divisor of T that fits ws, capped at 16
  // so the chunk activation buffer (+W1 +mem1) stays resident in the 192MB L2.
  int Tc = 1;
  const int cands[5] = {16, 8, 4, 2, 1};
  for (int i = 0; i < 5; ++i) {
    size_t need = (state_elems + (size_t)cands[i] * per_step) * sizeof(float);
    if (need <= ws_size) { Tc = cands[i]; break; }
  }
  float* cur2 = cur1 + (size_t)Tc * Bb * NH;      // Tc*B*NO

  // Zero persistent state (mem1, mem2 are adjacent at the base of ws).
  {
    size_t n = state_elems;
    int blocks = (int)((n + 255) / 256);
    zero_kernel<<<blocks, 256, 0, stream>>>((float*)d_ws, n);
  }

  for (int t0 = 0; t0 < T; t0 += Tc) {
    const int Tcur = (T - t0 < Tc) ? (T - t0) : Tc;
    const int Mrows = Tcur * Bb;

    // cur1 = x[t0:t0+Tcur] @ W1^T      (Mrows x NH, K = NI)
    gemm_nt_128x128<<<dim3(NH / 128, Mrows / 128), 256, 0, stream>>>(
        x + (size_t)t0 * Bb * NI, W1, cur1, Mrows, NH, NI);

    // LIF layer 1: cur1 -> spk1 (in place), carry mem1, emit traces.
    lif_scan1<<<dim3(NH / 256, Bb), 256, 0, stream>>>(
        cur1, b1, mem1, out_spk1, out_mem1, Tcur, t0, Bb, NH);

    // cur2 = spk1 @ W2^T               (Mrows x NO, K = NH)
    gemm_nt_128x128<<<dim3(NO / 128, Mrows / 128), 256, 0, stream>>>(
        cur1, W2, cur2, Mrows, NO, NH);

    // LIF layer 2: spk2 straight into d_out.
    lif_scan2<<<dim3((Bb * NO) / 256), 256, 0, stream>>>(
        cur2, b2, mem2, out_spk2, Tcur, t0, Bb, NO);
  }
}